// modulation_layer_23837068493438
// MI455X (gfx1250) — compile-verified
//
#include <hip/hip_runtime.h>
#include <hip/hip_bf16.h>

// ---------------------------------------------------------------------------
// out[b,i,j] = (u_real + i*u_imag) * exp(i*pi*2*sigmoid(phase[i,j]))
// u_*: (8, 2048, 2048) fp32, phase: (2048, 2048) fp32, out: complex64 interleaved.
// Memory-bound (~554 MB @ 23.3 TB/s ~= 24 us). Strategy:
//   * block owns one 1024-elem segment of the plane, loops over the 8 batches
//     -> cos/sin modulation computed once, register-resident, reused 8x
//   * u tiles streamed via CDNA5 async global->LDS B128 loads, double-buffered,
//     synchronized with s_wait_asynccnt (ASYNCcnt path)
//   * output written with non-temporal B128 stores (don't thrash L2)
// ---------------------------------------------------------------------------

typedef float f4 __attribute__((ext_vector_type(4)));

#define MESH   2048
#define BATCH  8
#define TPB    256
#define EPT    4                 // elements (floats) per thread = 16 bytes
#define TILE   (TPB * EPT)       // 1024 elements per block segment

// Async copy of 16 bytes: global (per-lane addr) -> LDS (per-lane 32-bit offset).
// Tracked by ASYNCcnt (CDNA5 ISA).
__device__ __forceinline__ void async_ld16(uint32_t lds_off, const float* gaddr) {
    asm volatile("global_load_async_to_lds_b128 %0, %1, off"
                 :: "v"(lds_off), "v"(gaddr)
                 : "memory");
}

__global__ __launch_bounds__(TPB) void modulation_kernel(
    const float* __restrict__ u_real,
    const float* __restrict__ u_imag,
    const float* __restrict__ phase_values,
    float* __restrict__ out)           // interleaved (re, im) float pairs
{
    __shared__ float lds_ur[2][TILE];
    __shared__ float lds_ui[2][TILE];

    const int    t     = threadIdx.x;
    const size_t base  = (size_t)blockIdx.x * TILE + (size_t)t * EPT;  // offset in plane
    const size_t plane = (size_t)MESH * (size_t)MESH;

    // LDS byte offsets of this thread's 16B slots (low 32 bits of the flat
    // pointer to a __shared__ object are the LDS offset — ISA §10.2 apertures).
    uint32_t l_ur[2], l_ui[2];
    l_ur[0] = (uint32_t)(uintptr_t)(&lds_ur[0][t * EPT]);
    l_ur[1] = (uint32_t)(uintptr_t)(&lds_ur[1][t * EPT]);
    l_ui[0] = (uint32_t)(uintptr_t)(&lds_ui[0][t * EPT]);
    l_ui[1] = (uint32_t)(uintptr_t)(&lds_ui[1][t * EPT]);

    // Kick off batch 0 tile fetch (2 async ops -> ASYNCcnt = 2).
    async_ld16(l_ur[0], u_real + base);
    async_ld16(l_ui[0], u_imag + base);

    // Overlap: compute register-resident modulation while batch-0 loads fly.
    // exp(i*pi*phase), phase = 2*sigmoid(pv)  ->  sincospif(2*sigmoid(pv))
    const f4 pv = *(const f4*)(phase_values + base);
    float c[EPT], s[EPT];
#pragma unroll
    for (int k = 0; k < EPT; ++k) {
        const float x   = (k == 0) ? pv.x : (k == 1) ? pv.y : (k == 2) ? pv.z : pv.w;
        const float sig = 1.0f / (1.0f + expf(-x));      // precise sigmoid
        sincospif(2.0f * sig, &s[k], &c[k]);             // sin/cos(pi * 2*sigmoid)
    }

#pragma unroll
    for (int b = 0; b < BATCH; ++b) {
        const int cur = b & 1;

        if (b + 1 < BATCH) {
            const int    nxt   = (b + 1) & 1;
            const size_t nbase = (size_t)(b + 1) * plane + base;
            async_ld16(l_ur[nxt], u_real + nbase);
            async_ld16(l_ui[nxt], u_imag + nbase);
            // All but the 2 just-issued (next-batch) loads complete -> batch b landed.
            asm volatile("s_wait_asynccnt 0x2" ::: "memory");
        } else {
            asm volatile("s_wait_asynccnt 0x0" ::: "memory");
        }

        // Each thread reads only the slot it staged itself -> per-wave wait suffices.
        const f4 ur = *(const f4*)(&lds_ur[cur][t * EPT]);
        const f4 ui = *(const f4*)(&lds_ui[cur][t * EPT]);

        f4 o0, o1;
        o0.x = ur.x * c[0] - ui.x * s[0];  o0.y = ur.x * s[0] + ui.x * c[0];
        o0.z = ur.y * c[1] - ui.y * s[1];  o0.w = ur.y * s[1] + ui.y * c[1];
        o1.x = ur.z * c[2] - ui.z * s[2];  o1.y = ur.z * s[2] + ui.z * c[2];
        o1.z = ur.w * c[3] - ui.w * s[3];  o1.w = ur.w * s[3] + ui.w * c[3];

        float* op = out + 2u * ((size_t)b * plane + base);
        __builtin_nontemporal_store(o0, (f4*)op);        // streaming store, skip L2 rinse
        __builtin_nontemporal_store(o1, (f4*)(op + 4));
    }
}

extern "C" void kernel_launch(void* const* d_in, const int* in_sizes, int n_in,
                              void* d_out, int out_size, void* d_ws, size_t ws_size,
                              hipStream_t stream) {
    const float* u_real = (const float*)d_in[0];
    const float* u_imag = (const float*)d_in[1];
    const float* phase  = (const float*)d_in[2];
    float*       out    = (float*)d_out;

    const int nblocks = (MESH * MESH) / TILE;   // 4096, exact fit
    modulation_kernel<<<dim3(nblocks), dim3(TPB), 0, stream>>>(u_real, u_imag, phase, out);
    (void)in_sizes; (void)n_in; (void)out_size; (void)d_ws; (void)ws_size;
}